// ConvCapsLayer_30726196035849
// MI455X (gfx1250) — compile-verified
//
#include <hip/hip_runtime.h>
#include <hip/hip_bf16.h>

typedef float v2f __attribute__((ext_vector_type(2)));
typedef float v8f __attribute__((ext_vector_type(8)));

#define NPATCH 450      // 2 * 15 * 15 patches
#define KKA    288      // K*K*A input capsules
#define EPS    1e-8f

// sum over the 4-lane group {4g..4g+3}; every lane ends with the group sum
__device__ __forceinline__ float groupsum4(float t) {
  t += __shfl_xor(t, 1, 32);
  t += __shfl_xor(t, 2, 32);
  return t;
}

__device__ __forceinline__ v8f pose_wmma(const float* __restrict__ xs,
                                         const float* __restrict__ wrow,
                                         int i, int aoff) {
  v2f av = *(const v2f*)(xs + i * 64 + aoff);
  const float* wp = wrow + (long)i * 512;
  v2f bv; bv.x = wp[0]; bv.y = wp[4];
  v8f cz = {};
  return __builtin_amdgcn_wmma_f32_16x16x4_f32(
      false, av, false, bv, (short)0, cz, false, false);
}

// One block = 4 waves, 4 patches. Wave w handles output capsules
// jbase..jbase+3 (jbase = blockIdx.y*16 + w*4). Each WMMA (f32 16x16x4)
// computes v[i] for the (4 patch x 4 pose-row) x (4 caps x 4 pose-col) tile.
__global__ __launch_bounds__(128, 1) void capsules_route(
    const float* __restrict__ x,     // (2,32,32,512)
    const float* __restrict__ wgt,   // (288,32,4,4)
    float* __restrict__ out) {       // (450,32,16) flat
  extern __shared__ float smem[];
  float* xs = smem;                                   // 4*288*16 = 18432 f
  const int tid  = threadIdx.x;
  const int wave = tid >> 5;
  const int lane = tid & 31;
  float* lg    = smem + 18432 + wave * (KKA * 16);    // raw logits, per wave
  float* stats = smem + 18432 + 4 * (KKA * 16) + wave * 32; // (max,1/den) x16

  const int nbase = blockIdx.x * 4;
  const int jbase = blockIdx.y * 16 + wave * 4;

  // ---- stage the 4 patches' poses into LDS: xs[(i*4+n4)*16 + p*4 + r] ----
  for (int t = tid; t < 4608; t += 128) {
    const int row = t >> 2, c4 = t & 3;       // row = i*4+n4, c4 = float4 idx
    const int i = row >> 2, n4 = row & 3;
    const int n = nbase + n4;
    float4 val = make_float4(0.f, 0.f, 0.f, 0.f);
    if (n < NPATCH) {
      const int b  = n / 225, rr = n % 225;
      const int oy = rr / 15, ox = rr % 15;
      const int ky = i / 96,  kxa = i % 96;
      const int kx = kxa >> 5, a = kxa & 31;
      const long off = ((long)((b * 32 + oy * 2 + ky) * 32 + (ox * 2 + kx))) * 512
                     + a * 16 + c4 * 4;
      val = *(const float4*)(x + off);
    }
    *(float4*)(xs + row * 16 + c4 * 4) = val;
  }
  __syncthreads();

  // ---- per-lane WMMA fragment addressing (CDNA5 layouts) ----
  // A (16x4 f32): lanes 0-15 hold M=0..15 with K=0,1; lanes 16-31 K=2,3.
  // B (4x16 f32): lanes 0-15 hold N=0..15 with K=0,1; lanes 16-31 K=2,3.
  // C (16x16):    VGPR k = row (k + 8*(lane>=16)), col = lane%16.
  const int M   = lane & 15;
  const int r0  = (lane < 16) ? 0 : 2;
  const int n4r = M >> 2;                 // patch for this A row
  const int pp  = M & 3;                  // pose row p
  const int j4  = (lane & 15) >> 2;       // capsule within wave's 4-tile
  const int qq  = lane & 3;               // pose col q
  const int aoff = n4r * 16 + pp * 4 + r0;                  // + i*64 (LDS)
  const float* wrow = wgt + (long)(jbase + j4) * 16 + r0 * 4 + qq; // + i*512
  // (n,j) pairs this lane's C registers touch: k<4 -> n4a, k>=4 -> n4b
  const int n4a = (lane < 16) ? 0 : 2;
  const int n4b = n4a + 1;
  const int pairA = n4a * 4 + j4;
  const int pairB = n4b * 4 + j4;

  float pose[8];

  // Fully unroll the 3 routing iterations: every iter-dependent predicate
  // becomes compile-time, so the 288-long WMMA loops are branch-free.
#pragma unroll
  for (int iter = 0; iter < 3; ++iter) {
    // ================= S pass: s = sum_i r_i * v_i =================
    float sAcc[8];
#pragma unroll
    for (int k = 0; k < 8; ++k) sAcc[k] = 0.f;
    float invA, invB;
    if (iter == 0) {
      invA = 1.f / 288.f; invB = 1.f / 288.f;
      for (int i = 0; i < KKA; ++i) {          // uniform routing weights
        v8f v = pose_wmma(xs, wrow, i, aoff);
#pragma unroll
        for (int k = 0; k < 8; ++k) sAcc[k] += v[k];
      }
    } else {
      const float mA = stats[pairA * 2];
      const float mB = stats[pairB * 2];
      invA = stats[pairA * 2 + 1];
      invB = stats[pairB * 2 + 1];
      for (int i = 0; i < KKA; ++i) {
        v8f v = pose_wmma(xs, wrow, i, aoff);
        // un-normalized softmax weights; 1/den folded in afterwards
        const float wA = __expf(lg[i * 16 + pairA] - mA);
        const float wB = __expf(lg[i * 16 + pairB] - mB);
#pragma unroll
        for (int k = 0; k < 4; ++k) sAcc[k] += wA * v[k];
#pragma unroll
        for (int k = 4; k < 8; ++k) sAcc[k] += wB * v[k];
      }
    }
#pragma unroll
    for (int k = 0; k < 4; ++k) sAcc[k] *= invA;
#pragma unroll
    for (int k = 4; k < 8; ++k) sAcc[k] *= invB;

    // ================= squash =================
    float tA = sAcc[0]*sAcc[0] + sAcc[1]*sAcc[1] + sAcc[2]*sAcc[2] + sAcc[3]*sAcc[3];
    float tB = sAcc[4]*sAcc[4] + sAcc[5]*sAcc[5] + sAcc[6]*sAcc[6] + sAcc[7]*sAcc[7];
    tA = groupsum4(tA); tB = groupsum4(tB);
    const float fA = tA / (1.f + tA) * rsqrtf(tA + EPS);
    const float fB = tB / (1.f + tB) * rsqrtf(tB + EPS);
#pragma unroll
    for (int k = 0; k < 4; ++k) pose[k] = sAcc[k] * fA;
#pragma unroll
    for (int k = 4; k < 8; ++k) pose[k] = sAcc[k] * fB;

    if (iter < 2) {
      // ============ L pass: logits_i += v_i . pose ============
      for (int i = 0; i < KKA; ++i) {
        v8f v = pose_wmma(xs, wrow, i, aoff);
        float dA = v[0]*pose[0] + v[1]*pose[1] + v[2]*pose[2] + v[3]*pose[3];
        float dB = v[4]*pose[4] + v[5]*pose[5] + v[6]*pose[6] + v[7]*pose[7];
        dA = groupsum4(dA); dB = groupsum4(dB);
        if ((lane & 3) == 0) {
          if (iter == 0) {                 // compile-time: plain store
            lg[i * 16 + pairA] = dA;
            lg[i * 16 + pairB] = dB;
          } else {                         // accumulate across iterations
            lg[i * 16 + pairA] += dA;
            lg[i * 16 + pairB] += dB;
          }
        }
      }
      __syncthreads();
      // ============ softmax stats per (n,j) pair ============
      if (lane < 16) {
        float m = -1e30f;
        for (int i = 0; i < KKA; ++i) m = fmaxf(m, lg[i * 16 + lane]);
        float den = 0.f;
        for (int i = 0; i < KKA; ++i) den += __expf(lg[i * 16 + lane] - m);
        stats[lane * 2]     = m;
        stats[lane * 2 + 1] = 1.f / den;
      }
      __syncthreads();
    }
  }

  // ---- store final poses: out[n*512 + j*16 + p*4 + q] ----
  const int j = jbase + j4;
#pragma unroll
  for (int k = 0; k < 8; ++k) {
    const int nn = nbase + ((k < 4) ? n4a : n4b);
    if (nn < NPATCH) {
      out[(long)nn * 512 + j * 16 + (k & 3) * 4 + qq] = pose[k];
    }
  }
}

extern "C" void kernel_launch(void* const* d_in, const int* in_sizes, int n_in,
                              void* d_out, int out_size, void* d_ws, size_t ws_size,
                              hipStream_t stream) {
  (void)in_sizes; (void)n_in; (void)out_size; (void)d_ws; (void)ws_size;
  const float* x = (const float*)d_in[0];      // (2,32,32,512) f32
  const float* w = (const float*)d_in[1];      // (288,32,4,4) f32
  float* out = (float*)d_out;                  // (2,15,15,512) f32

  const size_t shmem = (size_t)(18432 + 4 * KKA * 16 + 4 * 32) * sizeof(float);
  (void)hipFuncSetAttribute((const void*)capsules_route,
                            hipFuncAttributeMaxDynamicSharedMemorySize,
                            (int)shmem);
  dim3 grid(113, 2, 1);    // 113 patch tiles of 4, 2 halves of the 32 capsules
  dim3 block(128, 1, 1);   // 4 wave32s, one 4-capsule tile each
  capsules_route<<<grid, block, shmem, stream>>>(x, w, out);
}